// EvolveGCNOModel_49529562857567
// MI455X (gfx1250) — compile-verified
//
#include <hip/hip_runtime.h>
#include <cstdint>
#include <cstddef>

#define FDIM 64

typedef __attribute__((ext_vector_type(2))) float v2f;
typedef __attribute__((ext_vector_type(8))) float v8f;

__device__ __forceinline__ float sigmoidf(float v) {
  return 1.0f / (1.0f + __expf(-v));
}

// ---------------- Kernel 1: GRU evolution of W (64x64, tiny) ----------------
__global__ __launch_bounds__(256) void k_evolve(
    const float* __restrict__ W, const float* __restrict__ Wi,
    const float* __restrict__ Wh, const float* __restrict__ bi,
    const float* __restrict__ bh, float* __restrict__ Wn) {
  int idx = blockIdx.x * blockDim.x + threadIdx.x;
  if (idx >= FDIM * FDIM) return;
  int m = idx >> 6;
  int c = idx & 63;
  const float* wrow = W + (size_t)m * FDIM;
  float gir = bi[c], giz = bi[FDIM + c], gin = bi[2 * FDIM + c];
  float ghr = bh[c], ghz = bh[FDIM + c], ghn = bh[2 * FDIM + c];
  const float* wir = Wi + (size_t)c * FDIM;
  const float* wiz = Wi + (size_t)(FDIM + c) * FDIM;
  const float* win = Wi + (size_t)(2 * FDIM + c) * FDIM;
  const float* whr = Wh + (size_t)c * FDIM;
  const float* whz = Wh + (size_t)(FDIM + c) * FDIM;
  const float* whn = Wh + (size_t)(2 * FDIM + c) * FDIM;
  for (int k = 0; k < FDIM; ++k) {
    float wv = wrow[k];
    gir = fmaf(wv, wir[k], gir);
    giz = fmaf(wv, wiz[k], giz);
    gin = fmaf(wv, win[k], gin);
    ghr = fmaf(wv, whr[k], ghr);
    ghz = fmaf(wv, whz[k], ghz);
    ghn = fmaf(wv, whn[k], ghn);
  }
  float r = sigmoidf(gir + ghr);
  float z = sigmoidf(giz + ghz);
  float t = tanhf(gin + r * ghn);
  Wn[idx] = (1.0f - z) * t + z * wrow[c];
}

// ---------------- Kernel 2: degree (self-loop = init 1.0) ----------------
__global__ __launch_bounds__(256) void k_deg_init(float* __restrict__ deg, int n) {
  int i = blockIdx.x * blockDim.x + threadIdx.x;
  if (i < n) deg[i] = 1.0f;  // self-loop weight
}

__global__ __launch_bounds__(256) void k_deg_edges(const long long* __restrict__ col,
                                                   const float* __restrict__ w,
                                                   float* __restrict__ deg, int e) {
  int i = blockIdx.x * blockDim.x + threadIdx.x;
  if (i < e) atomicAdd(deg + col[i], w[i]);
}

__global__ __launch_bounds__(256) void k_dinv(const float* __restrict__ deg,
                                              float* __restrict__ dinv, int n) {
  int i = blockIdx.x * blockDim.x + threadIdx.x;
  if (i < n) {
    float d = deg[i];
    dinv[i] = (d > 0.0f) ? rsqrtf(d) : 0.0f;
  }
}

// ---------------- Kernel 3: h = x @ W via V_WMMA_F32_16X16X4_F32 ----------------
// 256 threads = 8 waves; block covers a 32(row) x 64(col) output tile:
// wave -> (rowTile = wave>>2, colTile = wave&3), each wave owns one 16x16 tile.
__global__ __launch_bounds__(256) void k_gemm(const float* __restrict__ x,
                                              const float* __restrict__ Wn,
                                              float* __restrict__ h, int n) {
  __shared__ float Ws[FDIM * FDIM];
  for (int i = threadIdx.x; i < FDIM * FDIM; i += 256) Ws[i] = Wn[i];
  __syncthreads();

  const int wave = threadIdx.x >> 5;
  const int lane = threadIdx.x & 31;
  const int hi = lane >> 4;  // 0 or 1 (lane group)
  const int l = lane & 15;

  const int m0 = (blockIdx.x * 2 + (wave >> 2)) * 16;  // output row base
  const int n0 = (wave & 3) * 16;                      // output col base

  int arow = m0 + l;
  if (arow >= n) arow = n - 1;  // clamp (N divisible by 32, so normally unused)
  const float* xrow = x + (size_t)arow * FDIM;

  v8f acc = {};
  for (int k0 = 0; k0 < FDIM; k0 += 4) {
    const int ka = k0 + hi * 2;
    v2f a, b;
    // A 16x4 f32: lanes 0-15 carry K=k0,k0+1; lanes 16-31 carry K=k0+2,k0+3
    a.x = xrow[ka];
    a.y = xrow[ka + 1];
    // B 4x16 f32: lane = N column; same K split across lane halves
    b.x = Ws[ka * FDIM + n0 + l];
    b.y = Ws[(ka + 1) * FDIM + n0 + l];
    acc = __builtin_amdgcn_wmma_f32_16x16x4_f32(false, a, false, b, (short)0,
                                                acc, false, false);
  }
  // C/D layout: VGPR v -> row m0 + v + 8*hi, col n0 + l
#pragma unroll
  for (int v = 0; v < 8; ++v) {
    int r = m0 + v + hi * 8;
    if (r < n) h[(size_t)r * FDIM + n0 + l] = acc[v];
  }
}

// ---------------- Kernel 4: agg init = self-loop message h[n] * dinv[n]^2 ----------------
__global__ __launch_bounds__(256) void k_agg_init(const float* __restrict__ h,
                                                  const float* __restrict__ dinv,
                                                  float* __restrict__ agg, int n) {
  int i = blockIdx.x * blockDim.x + threadIdx.x;
  if (i < n * FDIM) {
    float s = dinv[i >> 6];
    agg[i] = h[i] * s * s;
  }
}

// ---------------- Kernel 5: edge gather + atomic scatter-add ----------------
// One wave32 per edge; each lane moves 2 of the 64 features (256B per edge each way).
__global__ __launch_bounds__(256) void k_edge_msg(const long long* __restrict__ row,
                                                  const long long* __restrict__ col,
                                                  const float* __restrict__ w,
                                                  const float* __restrict__ dinv,
                                                  const float* __restrict__ h,
                                                  float* __restrict__ agg, int e) {
  int ed = blockIdx.x * 8 + (threadIdx.x >> 5);
  if (ed >= e) return;
  int lane = threadIdx.x & 31;
  long long r = row[ed];
  long long c = col[ed];
  float norm = dinv[r] * w[ed] * dinv[c];
  const float2 hv = *(const float2*)(h + (size_t)r * FDIM + lane * 2);
  float* op = agg + (size_t)c * FDIM + lane * 2;
  atomicAdd(op, hv.x * norm);
  atomicAdd(op + 1, hv.y * norm);
}

// ---------------- Kernel 6: head  y = relu(agg) @ fc_w^T + fc_b ----------------
__global__ __launch_bounds__(256) void k_head(const float* __restrict__ agg,
                                              const float* __restrict__ fcw,
                                              const float* __restrict__ fcb,
                                              float* __restrict__ y, int n) {
  int node = blockIdx.x * 8 + (threadIdx.x >> 5);
  if (node >= n) return;
  int lane = threadIdx.x & 31;
  const float* ap = agg + (size_t)node * FDIM;
  float s = fmaxf(ap[lane], 0.0f) * fcw[lane] +
            fmaxf(ap[lane + 32], 0.0f) * fcw[lane + 32];
  for (int off = 16; off > 0; off >>= 1) s += __shfl_xor(s, off, 32);
  if (lane == 0) y[node] = s + fcb[0];
}

extern "C" void kernel_launch(void* const* d_in, const int* in_sizes, int n_in,
                              void* d_out, int out_size, void* d_ws, size_t ws_size,
                              hipStream_t stream) {
  const float* x = (const float*)d_in[0];
  const long long* ei = (const long long*)d_in[1];  // int64 per reference
  const float* ew = (const float*)d_in[2];
  const float* W0 = (const float*)d_in[3];
  const float* Wi = (const float*)d_in[4];
  const float* Wh = (const float*)d_in[5];
  const float* bi = (const float*)d_in[6];
  const float* bh = (const float*)d_in[7];
  const float* fcw = (const float*)d_in[8];
  const float* fcb = (const float*)d_in[9];
  float* y = (float*)d_out;

  const int n = in_sizes[0] / FDIM;  // 100000
  const int e = in_sizes[1] / 2;     // 1600000
  const long long* row = ei;
  const long long* col = ei + e;

  // Workspace carve-up (all float32)
  float* ws = (float*)d_ws;
  float* Wn = ws;                      // 64*64
  float* deg = Wn + FDIM * FDIM;       // n
  float* dinv = deg + n;               // n
  float* h = dinv + n;                 // n*64
  float* agg = h + (size_t)n * FDIM;   // n*64

  k_evolve<<<(FDIM * FDIM + 255) / 256, 256, 0, stream>>>(W0, Wi, Wh, bi, bh, Wn);
  k_deg_init<<<(n + 255) / 256, 256, 0, stream>>>(deg, n);
  k_deg_edges<<<(e + 255) / 256, 256, 0, stream>>>(col, ew, deg, e);
  k_dinv<<<(n + 255) / 256, 256, 0, stream>>>(deg, dinv, n);
  k_gemm<<<(n + 31) / 32, 256, 0, stream>>>(x, Wn, h, n);
  k_agg_init<<<((size_t)n * FDIM + 255) / 256, 256, 0, stream>>>(h, dinv, agg, n);
  k_edge_msg<<<(e + 7) / 8, 256, 0, stream>>>(row, col, ew, dinv, h, agg, e);
  k_head<<<(n + 7) / 8, 256, 0, stream>>>(agg, fcw, fcb, y, n);
}